// RecurrentConstitutiveModel_22436909154895
// MI455X (gfx1250) — compile-verified
//
#include <hip/hip_runtime.h>

// J2 plasticity return-mapping scan: B=16384 paths, T=256 steps, C=6 Voigt.
// One thread per path; time chunks double-buffered through LDS via CDNA5
// async global->LDS DMA (ASYNCcnt). Memory-bound: ~201 MB traffic, ~8.6us
// floor at 23.3 TB/s.

#define kT    256
#define kC    6
#define kCH   8                 // time steps per chunk
#define kNCH  (kT / kCH)        // 32 chunks
#define kRowF 52                // padded LDS row: 48 data floats + 4 pad (208B, 16B aligned)
#define kSegs 12                // 16B segments of real data per row (192B)
#define kTPB  128               // threads per block == paths per block
#define SQ23  0.816496580927726f

typedef __attribute__((__vector_size__(16))) int v4i;
typedef __attribute__((address_space(1))) v4i* g4p;   // global v4i*
typedef __attribute__((address_space(3))) v4i* l4p;   // LDS v4i*

__device__ __forceinline__ void async_ld_b128(const float* g, float* l) {
#if __has_builtin(__builtin_amdgcn_global_load_async_to_lds_b128)
  __builtin_amdgcn_global_load_async_to_lds_b128(
      (g4p)(unsigned long long)g,
      (l4p)(unsigned)(unsigned long long)l, 0, 0);
#else
  asm volatile("global_load_async_to_lds_b128 %0, %1, off"
               :: "v"((unsigned)(unsigned long long)l), "v"(g)
               : "memory");
#endif
}

#if __has_builtin(__builtin_amdgcn_s_wait_asynccnt)
#define WAIT_ASYNC0() __builtin_amdgcn_s_wait_asynccnt(0)
#else
#define WAIT_ASYNC0() asm volatile("s_wait_asynccnt 0" ::: "memory")
#endif

__global__ __launch_bounds__(kTPB)
void j2_return_map_scan(const float* __restrict__ strain,
                        const float* __restrict__ Ep,
                        const float* __restrict__ nup,
                        const float* __restrict__ syp,
                        const float* __restrict__ Hp,
                        float* __restrict__ out) {
  __shared__ __align__(16) float smem[2][kTPB * kRowF];

  const int tid = threadIdx.x;
  const long pb = (long)blockIdx.x * kTPB;

  // uniform material parameters -> SGPRs
  const float E  = Ep[0];
  const float nu = nup[0];
  const float sy = syp[0];
  const float H  = Hp[0];
  const float mu       = E / (2.0f * (1.0f + nu));
  const float Kb       = E / (3.0f * (1.0f - 2.0f * nu));
  const float two_mu   = 2.0f * mu;
  const float inv_hard = 1.0f / (two_mu + (2.0f / 3.0f) * H);

  const float* gin  = strain + pb * (kT * kC);
  float*       gout = out    + pb * (kT * kC);

  // stage chunk `ch` of all kTPB paths into LDS buffer `b` (coalesced:
  // consecutive lanes copy consecutive 16B global segments)
  auto stage = [&](int ch, int b) {
    const int t0 = ch * kCH;
#pragma unroll
    for (int k = 0; k < kSegs; ++k) {
      const int s = tid + kTPB * k;       // 0 .. kTPB*12-1
      const int p = s / kSegs;            // path within block
      const int o = s - p * kSegs;        // 16B segment within path-chunk
      async_ld_b128(gin + p * (kT * kC) + t0 * kC + o * 4,
                    &smem[b][p * kRowF + o * 4]);
    }
  };

  // carried recurrence state (per path, in registers)
  float ep0 = 0.f, ep1 = 0.f, ep2 = 0.f, ep3 = 0.f, ep4 = 0.f, ep5 = 0.f;
  float alpha = 0.f;

  stage(0, 0);
  WAIT_ASYNC0();
  __syncthreads();

  for (int ch = 0; ch < kNCH; ++ch) {
    const int buf = ch & 1;
    if (ch + 1 < kNCH) stage(ch + 1, buf ^ 1);   // DMA overlaps compute below

    const float* row = &smem[buf][tid * kRowF];
    __align__(16) float sig[kCH * kC];

#pragma unroll
    for (int i = 0; i < kCH; ++i) {
      const float e0 = row[i * 6 + 0];
      const float e1 = row[i * 6 + 1];
      const float e2 = row[i * 6 + 2];
      const float e3 = row[i * 6 + 3];
      const float e4 = row[i * 6 + 4];
      const float e5 = row[i * 6 + 5];

      const float treps = e0 + e1 + e2;
      const float q0 = e0 - ep0, q1 = e1 - ep1, q2 = e2 - ep2;
      const float q3 = e3 - ep3, q4 = e4 - ep4, q5 = e5 - ep5;
      const float d  = (q0 + q1 + q2) * (1.0f / 3.0f);

      const float s0 = two_mu * (q0 - d);
      const float s1 = two_mu * (q1 - d);
      const float s2 = two_mu * (q2 - d);
      const float s3 = two_mu * q3;
      const float s4 = two_mu * q4;
      const float s5 = two_mu * q5;

      const float sn2 = s0 * s0 + s1 * s1 + s2 * s2 +
                        2.0f * (s3 * s3 + s4 * s4 + s5 * s5) + 1e-20f;
      const float snorm = sqrtf(sn2);

      const float f  = snorm - SQ23 * (sy + H * alpha);
      const float dg = fmaxf(f, 0.0f) * inv_hard;   // radial-return multiplier
      const float r  = dg / snorm;                  // dgamma / ||s_tr||

      ep0 += r * s0; ep1 += r * s1; ep2 += r * s2;
      ep3 += r * s3; ep4 += r * s4; ep5 += r * s5;
      alpha += SQ23 * dg;

      const float coef = 1.0f - two_mu * r;         // s = s_tr * coef
      const float kt   = Kb * treps;
      sig[i * 6 + 0] = s0 * coef + kt;
      sig[i * 6 + 1] = s1 * coef + kt;
      sig[i * 6 + 2] = s2 * coef + kt;
      sig[i * 6 + 3] = s3 * coef;
      sig[i * 6 + 4] = s4 * coef;
      sig[i * 6 + 5] = s5 * coef;
    }

    // write this path's chunk (192B, 16B-aligned) with 12 b128 stores
    float4* dst = (float4*)(gout + (size_t)tid * (kT * kC) + (size_t)ch * (kCH * kC));
    const float4* src = (const float4*)sig;
#pragma unroll
    for (int k = 0; k < kSegs; ++k) dst[k] = src[k];

    WAIT_ASYNC0();       // next chunk's DMA landed (this wave's portion)
    __syncthreads();     // all waves landed + everyone done reading `buf`
  }
}

extern "C" void kernel_launch(void* const* d_in, const int* in_sizes, int n_in,
                              void* d_out, int out_size, void* d_ws, size_t ws_size,
                              hipStream_t stream) {
  const float* strain = (const float*)d_in[0];
  const float* E  = (const float*)d_in[1];
  const float* nu = (const float*)d_in[2];
  const float* sy = (const float*)d_in[3];
  const float* H  = (const float*)d_in[4];
  float* out = (float*)d_out;

  const int nPaths = in_sizes[0] / (kT * kC);   // 16384
  dim3 grid(nPaths / kTPB), block(kTPB);
  hipLaunchKernelGGL(j2_return_map_scan, grid, block, 0, stream,
                     strain, E, nu, sy, H, out);
}